// SingleT2FLS_Mamdani_27530740367459
// MI455X (gfx1250) — compile-verified
//
#include <hip/hip_runtime.h>
#include <hip/hip_bf16.h>

// Problem constants (from reference)
#define RR 64      // rules
#define AA 16      // antecedents
#define NSAMP 16384
#define NT 64      // threads per block (1 sample per thread), 2 waves
#define DSTR 72    // padded LDS stride for d arrays (avoids half-wave bank conflicts)

typedef float v2f __attribute__((ext_vector_type(2)));
typedef float v8f __attribute__((ext_vector_type(8)));

// workspace float layout
#define WS_INV 0
#define WS_M1  1024
#define WS_M2  2048
#define WS_MM  3072
#define WS_CS1 4096
#define WS_CS2 4160
#define WS_IP  4224   // then: int invp1[64], invp2[64]

// ---------------------------------------------------------------------------
// Setup: precompute per-(r,a) membership params; stable rank-sort c1/c2.
// ---------------------------------------------------------------------------
__global__ __launch_bounds__(1024) void t2fls_setup(const float* __restrict__ w,
                                                    const float* __restrict__ c1,
                                                    const float* __restrict__ c2,
                                                    float* __restrict__ wsf) {
  const int tid = threadIdx.x;
  if (tid < RR * AA) {
    // locs[r,a] = 16r + a == tid; sigma=w[i], M1=w[i+1], M2=w[i+2]
    float sigma = w[tid] + 1e-4f;
    float M1 = w[tid + 1], M2 = w[tid + 2];
    float m1 = fminf(M1, M2), m2 = fmaxf(M1, M2);
    wsf[WS_INV + tid] = 1.0f / (2.0f * sigma * sigma);
    wsf[WS_M1 + tid] = m1;
    wsf[WS_M2 + tid] = m2;
    wsf[WS_MM + tid] = 0.5f * (m1 + m2);
  }
  if (tid < RR) {
    int* wsi = (int*)(wsf + WS_IP);
    float v1 = c1[tid], v2 = c2[tid];
    int r1 = 0, r2 = 0;
    for (int j = 0; j < RR; ++j) {   // stable rank (matches jnp.argsort stable)
      float u1 = c1[j], u2 = c2[j];
      r1 += (u1 < v1) || (u1 == v1 && j < tid);
      r2 += (u2 < v2) || (u2 == v2 && j < tid);
    }
    wsf[WS_CS1 + r1] = v1;    // sorted values
    wsf[WS_CS2 + r2] = v2;
    wsi[tid]      = r1;       // inverse permutation: rule -> rank
    wsi[RR + tid] = r2;
  }
}

// ---------------------------------------------------------------------------
// KM scan for one 16-sample tile via triangular-matmul WMMA (f32 16x16x4).
//   S[n,j]  = sum_{i<=j} cs[i]*d[n,i]   (acc in accS)
//   S1[n,j] = sum_{i<=j} d[n,i]         (acc in accS1)
// then per-row min/max of (s0+S)/(s10+S1) with j=-1 seed s0/s10.
// ---------------------------------------------------------------------------
template <bool ISMAX>
__device__ __forceinline__ void km_side(const float* __restrict__ sD,
                                        const float* __restrict__ sCS,
                                        const float* __restrict__ sS0,
                                        const float* __restrict__ sS10,
                                        float* __restrict__ sRes,
                                        int tb, int l) {
  const int half = l >> 4;        // which 16-lane half of the wave
  const int ln   = l & 15;
  v8f accS[4], accS1[4];
#pragma unroll
  for (int jt = 0; jt < 4; ++jt) {
    accS[jt]  = (v8f){0.f, 0.f, 0.f, 0.f, 0.f, 0.f, 0.f, 0.f};
    accS1[jt] = (v8f){0.f, 0.f, 0.f, 0.f, 0.f, 0.f, 0.f, 0.f};
  }
  for (int kk = 0; kk < RR; kk += 4) {
    // A-matrix 16x4 f32 layout: vgpr0 = K=kk(+2 for upper half), vgpr1 = K+1
    const int k0 = kk + 2 * half;
    const int k1 = k0 + 1;
    v2f a;
    a.x = sD[k0 * DSTR + tb + ln];
    a.y = sD[k1 * DSTR + tb + ln];
    const float cs0 = sCS[k0];
    const float cs1v = sCS[k1];
#pragma unroll
    for (int jt = 0; jt < 4; ++jt) {
      const int j = jt * 16 + ln;          // output column (rank index)
      v2f bcs, bone;
      bcs.x  = (k0 <= j) ? cs0  : 0.0f;    // B = triangular cs matrix, built on the fly
      bcs.y  = (k1 <= j) ? cs1v : 0.0f;
      bone.x = (k0 <= j) ? 1.0f : 0.0f;
      bone.y = (k1 <= j) ? 1.0f : 0.0f;
      accS[jt]  = __builtin_amdgcn_wmma_f32_16x16x4_f32(false, a, false, bcs,
                                                        (short)0, accS[jt], false, false);
      accS1[jt] = __builtin_amdgcn_wmma_f32_16x16x4_f32(false, a, false, bone,
                                                        (short)0, accS1[jt], false, false);
    }
  }
  // C/D layout: vgpr v holds row M=v (lanes 0-15) / M=v+8 (lanes 16-31), N=lane
#pragma unroll
  for (int v = 0; v < 8; ++v) {
    const int row = tb + v + 8 * half;     // sample index within block
    const float s0  = sS0[row];
    const float s10 = sS10[row];
    float best = __fdividef(s0, s10);      // j = -1 seed
#pragma unroll
    for (int jt = 0; jt < 4; ++jt) {
      float ratio = __fdividef(s0 + accS[jt][v], s10 + accS1[jt][v]);
      best = ISMAX ? fmaxf(best, ratio) : fminf(best, ratio);
    }
    // reduce over 16 columns held by the 16 lanes of this half
#pragma unroll
    for (int m = 8; m >= 1; m >>= 1) {
      float o = __shfl_xor(best, m, 32);
      best = ISMAX ? fmaxf(best, o) : fminf(best, o);
    }
    if (ln == 0) sRes[row] = best;
  }
}

// ---------------------------------------------------------------------------
// Main kernel: membership in log-domain (1 exp per (n,r) per bound instead of
// 16x2), then WMMA-based Karnik-Mendel type reduction.
// ---------------------------------------------------------------------------
__global__ __launch_bounds__(NT) void t2fls_main(const float* __restrict__ x,
                                                 const float* __restrict__ c1,
                                                 const float* __restrict__ c2,
                                                 const float* __restrict__ wsf,
                                                 float* __restrict__ out) {
  __shared__ float s_inv[RR * AA], s_m1[RR * AA], s_m2[RR * AA], s_mm[RR * AA];
  __shared__ float s_cs1[RR], s_cs2[RR], s_c1n[RR], s_c2n[RR];
  __shared__ int   s_ip1[RR], s_ip2[RR];
  __shared__ float s_dL[RR * DSTR], s_dR[RR * DSTR];
  __shared__ float s_s0L[NT], s_s10L[NT], s_s0R[NT], s_s10R[NT];
  __shared__ float s_resL[NT], s_resR[NT];

  const int tid  = threadIdx.x;
  const int samp = blockIdx.x * NT + tid;

  __builtin_prefetch(&x[samp * AA], 0, 1);   // global_prefetch_b8

  for (int i = tid; i < RR * AA; i += NT) {
    s_inv[i] = wsf[WS_INV + i];
    s_m1[i]  = wsf[WS_M1 + i];
    s_m2[i]  = wsf[WS_M2 + i];
    s_mm[i]  = wsf[WS_MM + i];
  }
  if (tid < RR) {
    s_cs1[tid] = wsf[WS_CS1 + tid];
    s_cs2[tid] = wsf[WS_CS2 + tid];
    s_c1n[tid] = c1[tid];
    s_c2n[tid] = c2[tid];
    const int* wsi = (const int*)(wsf + WS_IP);
    s_ip1[tid] = wsi[tid];
    s_ip2[tid] = wsi[RR + tid];
  }

  float xv[AA];
#pragma unroll
  for (int a = 0; a < AA; ++a) xv[a] = x[samp * AA + a];

  __syncthreads();

  float s0L = 0.f, s10L = 0.f, s0R = 0.f, s10R = 0.f;
  for (int r = 0; r < RR; ++r) {
    float su = 0.f, sl = 0.f;
#pragma unroll
    for (int a = 0; a < AA; ++a) {
      const int idx = r * AA + a;
      const float inv = s_inv[idx], m1 = s_m1[idx], m2 = s_m2[idx], mm = s_mm[idx];
      const float xa = xv[a];
      const float d1 = xa - m1, d2 = xa - m2;
      const float t1 = d1 * d1 * inv;      // -log g1
      const float t2 = d2 * d2 * inv;      // -log g2
      const float ut = (xa >= m1 && xa <= m2) ? 0.0f : ((xa > m2) ? t2 : t1);
      su -= ut;                            // log upper
      sl -= (xa <= mm) ? t2 : t1;          // log lower
    }
    const float U = __expf(su);            // prod of uppers
    const float L = __expf(sl);            // prod of lowers
    s0L = fmaf(s_c1n[r], L, s0L); s10L += L;
    s0R = fmaf(s_c2n[r], U, s0R); s10R += U;
    s_dL[s_ip1[r] * DSTR + tid] = U - L;   // d in c1-sorted order
    s_dR[s_ip2[r] * DSTR + tid] = L - U;   // d in c2-sorted order
  }
  s_s0L[tid] = s0L; s_s10L[tid] = s10L;
  s_s0R[tid] = s0R; s_s10R[tid] = s10R;
  __syncthreads();

  // Each wave (32 lanes) processes its 32 samples as two 16-row WMMA tiles.
  const int wave = tid >> 5;
  const int l    = tid & 31;
#pragma unroll
  for (int t = 0; t < 2; ++t) {
    const int tb = wave * 32 + t * 16;
    km_side<false>(s_dL, s_cs1, s_s0L, s_s10L, s_resL, tb, l);   // left: min
    km_side<true >(s_dR, s_cs2, s_s0R, s_s10R, s_resR, tb, l);   // right: max
  }
  __syncthreads();

  out[samp] = 0.5f * (s_resL[tid] + s_resR[tid]);
}

// ---------------------------------------------------------------------------
extern "C" void kernel_launch(void* const* d_in, const int* in_sizes, int n_in,
                              void* d_out, int out_size, void* d_ws, size_t ws_size,
                              hipStream_t stream) {
  const float* in = (const float*)d_in[0];   // (16384, 16) f32
  const float* w  = (const float*)d_in[1];   // (3072,) f32
  const float* c1 = (const float*)d_in[2];   // (64,) f32
  const float* c2 = (const float*)d_in[3];   // (64,) f32
  float* out = (float*)d_out;                // (16384,) f32
  float* wsf = (float*)d_ws;                 // uses 17 KB

  t2fls_setup<<<1, 1024, 0, stream>>>(w, c1, c2, wsf);
  t2fls_main<<<NSAMP / NT, NT, 0, stream>>>(in, c1, c2, wsf, out);
}